// FieldWeightedFactorizationMachineModel_86045374808106
// MI455X (gfx1250) — compile-verified
//
#include <hip/hip_runtime.h>

// FwFM: out[b] = w0 + sum_f bias[x[b,f]] + 0.5 * sum_ij (e_i . e_j) * W'_ij
// where W' = W with zero diagonal. Using Gram symmetry:
//   inter = 0.25 * sum_ij G_ij * S_ij,  S = W' + W'^T (symmetric, zero diag)
// Padded 39 fields -> 48 (3 tiles of 16). Only 6 of 9 tiles computed
// (diagonal tiles weight 1, upper tiles weight 2).

typedef float v2f __attribute__((ext_vector_type(2)));
typedef float v4f __attribute__((ext_vector_type(4)));
typedef float v8f __attribute__((ext_vector_type(8)));

#define NUM_FIELDS 39
#define EMBED_DIM  64
#define PF         48   // fields padded to 3 WMMA tiles
#define ESTRIDE    68   // LDS row stride in floats: bank = (4r+c)%64, conflict-free
#define WAVES      4
#define BLOCK      (WAVES * 32)

__global__ __launch_bounds__(BLOCK)
void fwfm_kernel(const int* __restrict__ x,
                 const float* __restrict__ emb,
                 const float* __restrict__ bias,
                 const float* __restrict__ W,
                 const float* __restrict__ w0,
                 float* __restrict__ out,
                 int batch)
{
    __shared__ float Elds[WAVES * PF * ESTRIDE];  // per-wave gathered embeddings
    __shared__ float Slds[PF * PF];               // symmetrized interaction weights

    const int tid  = threadIdx.x;
    const int wave = tid >> 5;
    const int lane = tid & 31;
    const int b    = blockIdx.x * WAVES + wave;
    const bool live = (b < batch);                // wave-uniform predicate

    // ---- Build S = W + W^T with zero diagonal, zero padding (cooperative) ----
    for (int i = tid; i < PF * PF; i += BLOCK) {
        int r = i / PF, c = i - r * PF;
        float v = 0.f;
        if (r < NUM_FIELDS && c < NUM_FIELDS && r != c)
            v = W[r * NUM_FIELDS + c] + W[c * NUM_FIELDS + r];
        Slds[i] = v;
    }

    float* Ew = &Elds[wave * PF * ESTRIDE];
    float acc_bias = 0.f;

    if (live) {
        const int* xb = x + (size_t)b * NUM_FIELDS;
        // ---- Gather 39 rows x 64 f32 via b128 loads into LDS ----
        for (int i = lane; i < NUM_FIELDS * (EMBED_DIM / 4); i += 32) {
            int r = i >> 4;            // 16 float4 chunks per row
            int c = (i & 15) << 2;
            const float* src = emb + (size_t)xb[r] * EMBED_DIM + c;
            v4f v = *(const v4f*)src;                 // global_load_b128
            *(v4f*)&Ew[r * ESTRIDE + c] = v;          // ds_store_b128
        }
        // zero-fill padded rows 39..47 (avoid NaN poisoning of padded Gram)
        for (int i = lane; i < (PF - NUM_FIELDS) * (EMBED_DIM / 4); i += 32) {
            int r = NUM_FIELDS + (i >> 4);
            int c = (i & 15) << 2;
            v4f z = {};
            *(v4f*)&Ew[r * ESTRIDE + c] = z;
        }
        // ---- bias gather (39 scalars across 32 lanes) ----
        if (lane < NUM_FIELDS)      acc_bias  = bias[xb[lane]];
        if (lane + 32 < NUM_FIELDS) acc_bias += bias[xb[lane + 32]];
    }

    __syncthreads();

    float acc = 0.f;
    if (live) {
        const int lm   = lane & 15;
        const int hi   = lane >> 4;    // half-select: K-offset for A/B, M-offset for C
        const int koff = hi * 2;

        const int   ti_arr[6]  = {0, 1, 2, 0, 0, 1};
        const int   tj_arr[6]  = {0, 1, 2, 1, 2, 2};
        const float wgt_arr[6] = {1.f, 1.f, 1.f, 2.f, 2.f, 2.f};

        #pragma unroll
        for (int t = 0; t < 6; ++t) {
            const int ti = ti_arr[t], tj = tj_arr[t];
            const float wgt = wgt_arr[t];
            // A(16x4) and B(4x16)=E^T fragments: identical contiguous float2 per lane
            const float* Arow = &Ew[(ti * 16 + lm) * ESTRIDE + koff];
            const float* Brow = &Ew[(tj * 16 + lm) * ESTRIDE + koff];
            v8f c = {};
            #pragma unroll
            for (int kb = 0; kb < EMBED_DIM; kb += 4) {
                v2f a  = *(const v2f*)(Arow + kb);    // ds_load_b64
                v2f bb = *(const v2f*)(Brow + kb);    // ds_load_b64
                c = __builtin_amdgcn_wmma_f32_16x16x4_f32(
                        false, a, false, bb, (short)0, c, false, false);
            }
            // Weighted reduction of this Gram tile against S
            const float* Srow = &Slds[(ti * 16 + hi * 8) * PF + tj * 16 + lm];
            #pragma unroll
            for (int vv = 0; vv < 8; ++vv)
                acc += wgt * c[vv] * Srow[vv * PF];
        }
    }

    // ---- wave32 reductions ----
    float tot  = acc;
    float btot = acc_bias;
    #pragma unroll
    for (int off = 16; off > 0; off >>= 1) {
        tot  += __shfl_xor(tot,  off, 32);
        btot += __shfl_xor(btot, off, 32);
    }
    if (live && lane == 0)
        out[b] = w0[0] + btot + 0.25f * tot;
}

extern "C" void kernel_launch(void* const* d_in, const int* in_sizes, int n_in,
                              void* d_out, int out_size, void* d_ws, size_t ws_size,
                              hipStream_t stream) {
    const int*   x    = (const int*)d_in[0];     // (16384, 39) int32
    const float* emb  = (const float*)d_in[1];   // (1e6, 64) f32
    const float* bias = (const float*)d_in[2];   // (1e6, 1)  f32
    const float* W    = (const float*)d_in[3];   // (39, 39)  f32
    const float* w0   = (const float*)d_in[4];   // (1,)      f32
    float* out = (float*)d_out;                  // (16384,)  f32

    const int batch  = out_size;
    const int blocks = (batch + WAVES - 1) / WAVES;
    fwfm_kernel<<<blocks, BLOCK, 0, stream>>>(x, emb, bias, W, w0, out, batch);
}